// SimilarityLoss_4629974745211
// MI455X (gfx1250) — compile-verified
//
#include <hip/hip_runtime.h>
#include <hip/hip_bf16.h>
#include <math.h>

// Problem constants (match reference)
#define B 64
#define N_WAY 5
#define N_SHOT 5
#define QUERY 15
#define NUM_SLOT 8
#define D 640
#define IMGS (N_WAY * QUERY)         // 75
#define LROWS (IMGS * N_WAY)         // 375 candidate rows per batch
#define NS (B * N_WAY * N_SHOT)      // 1600 support images

typedef __attribute__((ext_vector_type(2))) float v2f;
typedef __attribute__((ext_vector_type(8))) float v8f;

// ---------------- workspace layout (floats) ----------------
#define WS_SCORES 0        // [2560]
#define WS_M      2560     // [24000]
#define WS_DIST   26560    // [120000]
#define WS_ACCUM  146560   // [8]
#define WS_MAXS   146576   // [320] ints

__device__ __forceinline__ float wave_reduce_add(float v) {
  for (int off = 16; off > 0; off >>= 1) v += __shfl_xor(v, off, 32);
  return v;
}

// ---------------------------------------------------------------------------
// K1: scores[r, slot] = mean over (shot, d) of support features.
// grid = 320 blocks (r = b*N_WAY + c), 256 threads = 8 waves (wave = slot).
// ---------------------------------------------------------------------------
__global__ void scores_kernel(const float* __restrict__ out_f,
                              float* __restrict__ scores) {
  const int r    = blockIdx.x;            // 0..319
  const int slot = threadIdx.x >> 5;      // wave id 0..7
  const int lane = threadIdx.x & 31;
  const float* base = out_f + (size_t)r * (N_SHOT * NUM_SLOT * D) + slot * D;
  float s = 0.f;
  for (int shot = 0; shot < N_SHOT; ++shot) {
    const float* p = base + shot * (NUM_SLOT * D);
    for (int dd = lane; dd < D; dd += 32) s += p[dd];
  }
  s = wave_reduce_add(s);
  if (lane == 0) scores[r * NUM_SLOT + slot] = s * (1.0f / (N_SHOT * D));
}

// ---------------------------------------------------------------------------
// K2: greedy cover_select per batch; thread j handles batch j. Also zeroes
// the accumulators (runs before finalize on the same stream every launch).
// ---------------------------------------------------------------------------
__global__ void select_kernel(const float* __restrict__ scores,
                              int* __restrict__ maxs,
                              float* __restrict__ accum) {
  const int j = threadIdx.x;
  if (j == 0)
    for (int t = 0; t < 8; ++t) accum[t] = 0.f;
  if (j >= B) return;
  unsigned taken = 0;
  for (int c = 0; c < N_WAY; ++c) {
    const float* row = scores + (j * N_WAY + c) * NUM_SLOT;
    float best = -3.0e38f;
    int bs = 0;
    for (int s = 0; s < NUM_SLOT; ++s) {
      if (taken & (1u << s)) continue;       // == where(taken, -1e30) then argmax
      float v = row[s];
      if (v > best) { best = v; bs = s; }    // first-max tie-break
    }
    taken |= (1u << bs);
    maxs[j * N_WAY + c] = bs;
  }
}

// ---------------------------------------------------------------------------
// K3: main WMMA kernel. One block per batch (64 blocks x 256 threads = 8
// waves). Prototype raw data is staged to LDS with the gfx1250 async path
// (global_load_async_to_lds_b128 + s_wait_asynccnt), then each wave runs
// 16-row candidate tiles through V_WMMA_F32_16X16X4_F32 against the protos,
// producing dist[j,l,c] = |x|^2 - 2 x.p_c + |p_c|^2 and m[j,l] = mean(x).
// ---------------------------------------------------------------------------
__global__ void main_kernel(const float* __restrict__ out_f,
                            const int* __restrict__ maxs,
                            float* __restrict__ dist,
                            float* __restrict__ mws) {
  __shared__ float sProto[16 * D];          // 40 KB, rows 5..15 zero-padded
  __shared__ float sStage[N_SHOT * D];      // 12.8 KB async staging buffer
  __shared__ float sPnorm[16];
  __shared__ int   sMaxs[8];

  const int j    = blockIdx.x;
  const int tid  = threadIdx.x;
  const int wave = tid >> 5;
  const int lane = tid & 31;
  const int half = lane >> 4;        // k-subpair select
  const int mrow = lane & 15;        // A-matrix row / B-matrix column

  if (tid < N_WAY) sMaxs[tid] = maxs[j * N_WAY + tid];
  // zero pad rows 5..15 of the prototype tile
  for (int idx = N_WAY * D + tid; idx < 16 * D; idx += blockDim.x)
    sProto[idx] = 0.f;
  __syncthreads();

  // --- prototype build, one class at a time -------------------------------
  // Stage the 5 selected-slot shot rows (5 x 2560B) into LDS via the async
  // DMA path, then reduce to proto[c] = mean over shots.
  const unsigned stage_lds = (unsigned)(uintptr_t)(&sStage[0]);
  for (int c = 0; c < N_WAY; ++c) {
    const int slot = sMaxs[c];
    const float* cbase =
        out_f + (size_t)(j * N_WAY + c) * (N_SHOT * NUM_SLOT * D) + slot * D;
    // 5 rows * 160 chunks of 16B = 800 chunks; LDS offset == chunk*16.
    for (int q = tid; q < N_SHOT * (D / 4); q += blockDim.x) {
      const int shot = q / (D / 4);
      const int part = q - shot * (D / 4);            // 16B chunk within row
      const unsigned long long gaddr =
          (unsigned long long)(uintptr_t)(cbase + shot * (NUM_SLOT * D) + part * 4);
      const unsigned laddr = stage_lds + (unsigned)q * 16u;
      asm volatile("global_load_async_to_lds_b128 %0, %1, off"
                   :: "v"(laddr), "v"(gaddr)
                   : "memory");
    }
    asm volatile("s_wait_asynccnt 0x0" ::: "memory");
    __syncthreads();
    for (int dd = tid; dd < D; dd += blockDim.x) {
      float s = 0.f;
      for (int shot = 0; shot < N_SHOT; ++shot) s += sStage[shot * D + dd];
      sProto[c * D + dd] = s * (1.0f / N_SHOT);
    }
    __syncthreads();   // protect sStage before next class overwrites it
  }

  // Prototype squared norms (waves 0..4), zero-pad the rest.
  if (wave < N_WAY) {
    float s = 0.f;
    for (int dd = lane; dd < D; dd += 32) {
      float p = sProto[wave * D + dd];
      s += p * p;
    }
    s = wave_reduce_add(s);
    if (lane == 0) sPnorm[wave] = s;
  }
  if (tid >= N_WAY && tid < 16) sPnorm[tid] = 0.f;
  __syncthreads();

  const float* qbase = out_f + (size_t)NS * (NUM_SLOT * D);

  for (int t = wave; t < 24; t += 8) {          // 24 tiles of 16 rows = 384 >= 375
    const int  l     = t * 16 + mrow;
    const bool valid = (l < LROWS);
    const int  lc    = valid ? l : 0;
    const int  img   = lc / N_WAY;
    const int  ii    = lc - img * N_WAY;
    const int  slot  = sMaxs[ii];
    const float* rp =
        qbase + ((size_t)(j * IMGS + img) * NUM_SLOT + slot) * D;

    v8f acc = {};
    float sx = 0.f, sx2 = 0.f;
    const float* bp = sProto + mrow * D;

    // Explicitly batched k-loop: 8 A-frag + 8 B-frag loads issued up front,
    // then 8 WMMAs, so >=8 global loads stay in flight every iteration.
    for (int kk = 0; kk < D / 4; kk += 8) {
      v2f a[8], bfr[8];
#pragma unroll
      for (int u = 0; u < 8; ++u) {
        const int kb = (kk + u) * 4 + half * 2;
        a[u].x = rp[kb];
        a[u].y = rp[kb + 1];
      }
#pragma unroll
      for (int u = 0; u < 8; ++u) {
        const int kb = (kk + u) * 4 + half * 2;
        bfr[u].x = bp[kb];
        bfr[u].y = bp[kb + 1];
      }
#pragma unroll
      for (int u = 0; u < 8; ++u) {
        acc = __builtin_amdgcn_wmma_f32_16x16x4_f32(
            false, a[u], false, bfr[u], (short)0, acc, false, false);
        sx  += a[u].x + a[u].y;
        sx2 += a[u].x * a[u].x + a[u].y * a[u].y;
      }
    }
    // combine the two k-halves of each row
    sx  += __shfl_xor(sx, 16, 32);
    sx2 += __shfl_xor(sx2, 16, 32);

    // D-tile layout: acc[v] in lane L is G[M = v + 8*(L/16)][N = L%16]
    for (int v = 0; v < 8; ++v) {
      const int M  = v + half * 8;
      const float rn = __shfl(sx2, M, 32);      // row M's |x|^2 lives in lane M
      const float dv = rn - 2.0f * acc[v] + sPnorm[mrow];
      const int lr = t * 16 + M;
      if (mrow < N_WAY && lr < LROWS)
        dist[((size_t)j * LROWS + lr) * N_WAY + mrow] = dv;
    }
    if (half == 0 && valid)
      mws[(size_t)j * LROWS + l] = sx * (1.0f / D);
  }
}

// ---------------------------------------------------------------------------
// K4: per-batch finalize. Running prefix argmax over m (get_22 row pick),
// log-softmax NLL + accuracy for both row selections, atomic accumulate.
// ---------------------------------------------------------------------------
__global__ void finalize_kernel(const float* __restrict__ mws,
                                const float* __restrict__ dist,
                                const int* __restrict__ labels_q,
                                float* __restrict__ accum) {
  const int j = threadIdx.x;
  if (j >= B) return;
  const float* mj = mws + (size_t)j * LROWS;
  const float* dj = dist + (size_t)j * LROWS * N_WAY;

  float nll = 0.f, acc = 0.f, nllt = 0.f, acct = 0.f;
  float best = -3.0e38f;
  int bi = 0;
  for (int s = 0; s < IMGS; ++s) {
    for (int i2 = 0; i2 < N_WAY; ++i2) {        // extend prefix to (s+1)*5
      const float v = mj[s * N_WAY + i2];
      if (v > best) { best = v; bi = s * N_WAY + i2; }
    }
    // label indexing follows logits.reshape(B, QUERY, N_WAY, N_WAY)
    const int lbl = labels_q[(j * QUERY + s / N_WAY) * N_WAY + (s % N_WAY)];
    const int ls  = s * N_WAY + s / QUERY;      // out_query row -> cand flat idx

    const float* rows[2] = { dj + ls * N_WAY, dj + bi * N_WAY };
    float outn[2], outa[2];
    for (int p = 0; p < 2; ++p) {
      const float* r = rows[p];
      float dmin = r[0];
      int am = 0;
      for (int c = 1; c < N_WAY; ++c)
        if (r[c] < dmin) { dmin = r[c]; am = c; }   // argmax(logits) = first argmin(dist)
      float se = 0.f;
      for (int c = 0; c < N_WAY; ++c) se += expf(dmin - r[c]);
      outn[p] = (r[lbl] - dmin) + logf(se);         // -log_softmax(-d)[lbl]
      outa[p] = (am == lbl) ? 1.0f : 0.0f;
    }
    nll += outn[0]; acc += outa[0];
    nllt += outn[1]; acct += outa[1];
  }
  atomicAdd(&accum[0], nll);
  atomicAdd(&accum[1], acc);
  atomicAdd(&accum[2], nllt);
  atomicAdd(&accum[3], acct);
}

// K5: emit (loss, loss_t, acc, acc_t)
__global__ void emit_kernel(const float* __restrict__ accum,
                            const float* __restrict__ att,
                            float* __restrict__ out) {
  if (threadIdx.x == 0 && blockIdx.x == 0) {
    const float inv = 1.0f / (B * IMGS);
    out[0] = accum[0] * inv + 0.1f * att[0];   // loss (nll + lambda*att)
    out[1] = accum[2] * inv;                   // loss_t (nll_t only)
    out[2] = accum[1] * inv;                   // acc
    out[3] = accum[3] * inv;                   // acc_t
  }
}

extern "C" void kernel_launch(void* const* d_in, const int* in_sizes, int n_in,
                              void* d_out, int out_size, void* d_ws, size_t ws_size,
                              hipStream_t stream) {
  (void)in_sizes; (void)n_in; (void)out_size; (void)ws_size;
  const float* out_f    = (const float*)d_in[0];
  // d_in[1] = labels_support (unused by the reference)
  const int*   labels_q = (const int*)d_in[2];
  const float* att      = (const float*)d_in[3];
  float* out = (float*)d_out;

  float* W      = (float*)d_ws;
  float* scores = W + WS_SCORES;
  float* mws    = W + WS_M;
  float* dist   = W + WS_DIST;
  float* accum  = W + WS_ACCUM;
  int*   maxs   = (int*)(W + WS_MAXS);

  scores_kernel  <<<B * N_WAY, 256, 0, stream>>>(out_f, scores);
  select_kernel  <<<1, 64, 0, stream>>>(scores, maxs, accum);
  main_kernel    <<<B, 256, 0, stream>>>(out_f, maxs, dist, mws);
  finalize_kernel<<<1, 64, 0, stream>>>(mws, dist, labels_q, accum);
  emit_kernel    <<<1, 32, 0, stream>>>(accum, att, out);
}